// FPSSampler_1529008357473
// MI455X (gfx1250) — compile-verified
//
#include <hip/hip_runtime.h>
#include <cstdint>
#include <cstddef>

// Problem constants (match reference): x (B, C, N) f32 -> y (B, C, M) f32
#define FPS_B   8
#define FPS_C   3
#define FPS_N   65536
#define FPS_M   512
#define THREADS 1024                     // 32 wave32 waves per workgroup
#define VPT     4                        // points per thread per pass (float4)
#define PASSES  (FPS_N / (THREADS * VPT))  // 16

// Dynamic LDS layout (single extern shared block => base offset 0).
// CDNA5: 320KB LDS per WGP lets the whole dist[] array stay on-chip
// (impossible in CDNA4's 64KB LDS).
#define SM_DIST  0                          // FPS_N floats   = 262144 B
#define SM_SEL   (SM_DIST + FPS_N * 4)      // FPS_M ints     =   2048 B
#define SM_REDV  (SM_SEL  + FPS_M * 4)      // 32 floats
#define SM_REDI  (SM_REDV + 32 * 4)         // 32 ints
#define SM_CEN   (SM_REDI + 32 * 4)         // 4 floats (centroid, async-LDS target)
#define SM_FAR   (SM_CEN  + 4 * 4)          // 1 int
#define SM_BYTES (SM_FAR  + 4)              // ~264.5 KB < 320 KB

__global__ __launch_bounds__(THREADS, 1)
void fps_sample_kernel(const float* __restrict__ x, float* __restrict__ y) {
    extern __shared__ unsigned char smem[];
    float* dist = reinterpret_cast<float*>(smem + SM_DIST);
    int*   sel  = reinterpret_cast<int*>  (smem + SM_SEL);
    float* redv = reinterpret_cast<float*>(smem + SM_REDV);
    int*   redi = reinterpret_cast<int*>  (smem + SM_REDI);
    float* cen  = reinterpret_cast<float*>(smem + SM_CEN);
    int*   pfar = reinterpret_cast<int*>  (smem + SM_FAR);

    const int b    = blockIdx.x;
    const int tid  = threadIdx.x;
    const int lane = tid & 31;
    const int wave = tid >> 5;

    const float* __restrict__ px = x + (size_t)b * FPS_C * FPS_N;  // plane c=0
    const float* __restrict__ py = px + FPS_N;                     // plane c=1
    const float* __restrict__ pz = px + 2 * FPS_N;                 // plane c=2

    // dist = +inf (vectorized: ds_store_b128)
    {
        const float4 inf4 = make_float4(__builtin_inff(), __builtin_inff(),
                                        __builtin_inff(), __builtin_inff());
        for (int i = tid * VPT; i < FPS_N; i += THREADS * VPT)
            *reinterpret_cast<float4*>(dist + i) = inf4;
    }
    __syncthreads();

    int far = 0;  // reference starts from index 0
    for (int m = 0; m < FPS_M; ++m) {
        if (tid == 0) sel[m] = far;

        // Broadcast centroid coords global -> LDS via async-to-LDS (ASYNCcnt path).
        // 3 active lanes, one b32 transfer each.
        if (tid < FPS_C) {
            uint64_t gaddr = (uint64_t)(uintptr_t)(px + (size_t)tid * FPS_N + far);
            // low 32 bits of a generic pointer into LDS == wave-relative LDS offset
            uint32_t laddr = (uint32_t)(uintptr_t)(&cen[tid]);
            asm volatile("global_load_async_to_lds_b32 %0, %1, off"
                         :: "v"(laddr), "v"(gaddr)
                         : "memory");
        }
        asm volatile("s_wait_asynccnt 0x0" ::: "memory");
        __syncthreads();

        const float cx = cen[0], cy = cen[1], cz = cen[2];

        // dist-update + thread-local argmax.
        // Per-thread point indices are strictly increasing, so strict '>' keeps
        // the first (lowest-index) maximum like jnp.argmax.
        float best  = -1.0f;
        int   bestI = 0;
        #pragma unroll 4
        for (int p = 0; p < PASSES; ++p) {
            const int i0 = p * (THREADS * VPT) + tid * VPT;  // 16B-aligned
            const float4 vx = *reinterpret_cast<const float4*>(px + i0);
            const float4 vy = *reinterpret_cast<const float4*>(py + i0);
            const float4 vz = *reinterpret_cast<const float4*>(pz + i0);
            float4 vd = *reinterpret_cast<float4*>(dist + i0);  // ds_load_b128

            {   const float dx = vx.x - cx, dy = vy.x - cy, dz = vz.x - cz;
                const float d  = dx*dx + dy*dy + dz*dz;
                vd.x = fminf(vd.x, d);
                if (vd.x > best) { best = vd.x; bestI = i0; } }
            {   const float dx = vx.y - cx, dy = vy.y - cy, dz = vz.y - cz;
                const float d  = dx*dx + dy*dy + dz*dz;
                vd.y = fminf(vd.y, d);
                if (vd.y > best) { best = vd.y; bestI = i0 + 1; } }
            {   const float dx = vx.z - cx, dy = vy.z - cy, dz = vz.z - cz;
                const float d  = dx*dx + dy*dy + dz*dz;
                vd.z = fminf(vd.z, d);
                if (vd.z > best) { best = vd.z; bestI = i0 + 2; } }
            {   const float dx = vx.w - cx, dy = vy.w - cy, dz = vz.w - cz;
                const float d  = dx*dx + dy*dy + dz*dz;
                vd.w = fminf(vd.w, d);
                if (vd.w > best) { best = vd.w; bestI = i0 + 3; } }

            *reinterpret_cast<float4*>(dist + i0) = vd;          // ds_store_b128
        }

        // wave32 shuffle argmax (smaller index wins ties, matching jnp.argmax)
        #pragma unroll
        for (int off = 16; off > 0; off >>= 1) {
            const float ov = __shfl_down(best,  off, 32);
            const int   oi = __shfl_down(bestI, off, 32);
            if (ov > best || (ov == best && oi < bestI)) { best = ov; bestI = oi; }
        }
        if (lane == 0) { redv[wave] = best; redi[wave] = bestI; }
        __syncthreads();

        // cross-wave reduce in wave 0 (exactly 32 partials for 32 waves)
        if (wave == 0) {
            best  = redv[lane];
            bestI = redi[lane];
            #pragma unroll
            for (int off = 16; off > 0; off >>= 1) {
                const float ov = __shfl_down(best,  off, 32);
                const int   oi = __shfl_down(bestI, off, 32);
                if (ov > best || (ov == best && oi < bestI)) { best = ov; bestI = oi; }
            }
            if (lane == 0) *pfar = bestI;
        }
        __syncthreads();
        far = *pfar;
    }
    __syncthreads();

    // Gather: y[b, c, j] = x[b, c, sel[j]]
    for (int t = tid; t < FPS_C * FPS_M; t += THREADS) {
        const int c = t / FPS_M;
        const int j = t - c * FPS_M;
        y[(size_t)b * FPS_C * FPS_M + t] = px[(size_t)c * FPS_N + sel[j]];
    }
}

extern "C" void kernel_launch(void* const* d_in, const int* in_sizes, int n_in,
                              void* d_out, int out_size, void* d_ws, size_t ws_size,
                              hipStream_t stream) {
    (void)in_sizes; (void)n_in; (void)d_ws; (void)ws_size; (void)out_size;
    const float* x = (const float*)d_in[0];
    float*       y = (float*)d_out;
    fps_sample_kernel<<<dim3(FPS_B), dim3(THREADS), SM_BYTES, stream>>>(x, y);
}